// VectorQuantizer_1580547968299
// MI455X (gfx1250) — compile-verified
//
#include <hip/hip_runtime.h>

typedef float v2f __attribute__((ext_vector_type(2)));
typedef float v4f __attribute__((ext_vector_type(4)));
typedef float v8f __attribute__((ext_vector_type(8)));
typedef int   v4i __attribute__((__vector_size__(16)));   // matches builtin param type

#define N_TOK   65536
#define DIM     64
#define KCODES  1024
#define BETA    0.25f

#define WAVES           8
#define ROWS_PER_WAVE   16
#define ROWS_PER_BLOCK  (WAVES * ROWS_PER_WAVE)   // 128 rows / block
#define CHUNK           128                        // codes per LDS chunk (32 KB)
#define NCHUNK          (KCODES / CHUNK)           // 8
#define TILES           (CHUNK / 16)               // 8 x (16-code) WMMA tiles / chunk

#if __has_builtin(__builtin_amdgcn_global_load_async_to_lds_b128)
#define HAVE_ASYNC_LDS 1
#else
#define HAVE_ASYNC_LDS 0
#endif

typedef __attribute__((address_space(1))) v4i* gv4i_p;   // global int4*
typedef __attribute__((address_space(3))) v4i* lv4i_p;   // LDS int4*

__device__ __forceinline__ void wait_async_all() {
#if __has_builtin(__builtin_amdgcn_s_wait_asynccnt)
  __builtin_amdgcn_s_wait_asynccnt(0);
#elif HAVE_ASYNC_LDS
  asm volatile("s_wait_asynccnt 0x0" ::: "memory");
#endif
}

// Stage CHUNK (=128) codebook rows (32 KB) into LDS: 2048 x b128, 8 per thread.
// CDNA5 path: GLOBAL_LOAD_ASYNC_TO_LDS_B128 (ASYNCcnt), no VGPR round-trip.
__device__ __forceinline__ void stage_chunk(const float* __restrict__ embedding,
                                            float* ldsDst, int codeBase, int tid) {
  const v4f* src = (const v4f*)(embedding + codeBase * DIM);
  v4f* dst = (v4f*)ldsDst;
#pragma unroll
  for (int i = 0; i < 8; ++i) {
    const int idx = tid + i * 256;
#if HAVE_ASYNC_LDS
    __builtin_amdgcn_global_load_async_to_lds_b128(
        (gv4i_p)(src + idx),
        (lv4i_p)(dst + idx),
        /*offset=*/0, /*cpol=*/0);
#else
    dst[idx] = src[idx];
#endif
  }
}

// D = A x B + C with V_WMMA_F32_16X16X4_F32 (exact fp32, matches reference argmin).
// A (16x4): lane<16 holds row (lane&15), cols {4k,4k+1}; lane>=16 cols {4k+2,4k+3}.
// B (4x16): lane<16 holds col (lane&15), rows {4k,4k+1}; lane>=16 rows {4k+2,4k+3}.
// C/D (16x16): VGPR v -> M=v (lanes 0-15) / M=v+8 (lanes 16-31); N = lane&15.

__global__ __launch_bounds__(256) void vq_argmin_kernel(
    const float* __restrict__ latents,
    const float* __restrict__ embedding,
    float* __restrict__ out,
    float* __restrict__ blockSse)
{
  __shared__ float bufS[2][CHUNK * DIM];            // 2 x 32 KB double buffer
  __shared__ float cAll[KCODES];                    // 0.5*||e_k||^2, all 1024 codes
  __shared__ int   idxS[WAVES * ROWS_PER_WAVE];
  __shared__ float sseS[WAVES];

  const int tid  = threadIdx.x;
  const int lane = tid & 31;
  const int wave = tid >> 5;
  const int p    = lane >> 4;        // half-wave: which K-pair of the 16x16x4 step
  const int r    = lane & 15;       // row (A) / col (B) within tile
  const int rowBase = (blockIdx.x * WAVES + wave) * ROWS_PER_WAVE;

  // kick off async staging of chunk 0 while we do register/scalar setup
  stage_chunk(embedding, bufS[0], 0, tid);

  // ---- load A tile (16 rows x D=64) into 16 v2f regs, WMMA A layout ----
  v2f a[16];
  {
    const float* Arow = latents + (rowBase + r) * DIM + 2 * p;
#pragma unroll
    for (int kk = 0; kk < 16; ++kk)
      a[kk] = *(const v2f*)(Arow + 4 * kk);
  }

  // ---- one-time 0.5*||e_k||^2 for all 1024 codes (4 codes / thread) ----
#pragma unroll
  for (int i = 0; i < 4; ++i) {
    const int code = tid * 4 + i;
    const v4f* e = (const v4f*)(embedding + code * DIM);
    float s = 0.f;
#pragma unroll
    for (int j = 0; j < DIM / 4; ++j) {
      v4f x = e[j];
      s += x.x * x.x + x.y * x.y + x.z * x.z + x.w * x.w;
    }
    cAll[code] = 0.5f * s;
  }

  float bestVal[8];
  int   bestIdx[8];
#pragma unroll
  for (int v = 0; v < 8; ++v) { bestVal[v] = -3.402823466e38f; bestIdx[v] = 0; }

  wait_async_all();
  __syncthreads();                     // chunk 0 staged, cAll ready

  for (int ch = 0; ch < NCHUNK; ++ch) {
    const int codeBase = ch * CHUNK;
    const float* cur = bufS[ch & 1];

    // overlap: DMA next chunk into the other buffer while we compute
    if (ch + 1 < NCHUNK)
      stage_chunk(embedding, bufS[(ch + 1) & 1], (ch + 1) * CHUNK, tid);

    // two code tiles per pass -> two independent WMMA accumulation chains
    for (int t = 0; t < TILES; t += 2) {
      const float* B0 = cur + ((t + 0) * 16 + r) * DIM + 2 * p;
      const float* B1 = cur + ((t + 1) * 16 + r) * DIM + 2 * p;
      v2f b0[16], b1[16];
#pragma unroll
      for (int kk = 0; kk < 16; ++kk) {
        b0[kk] = *(const v2f*)(B0 + 4 * kk);
        b1[kk] = *(const v2f*)(B1 + 4 * kk);
      }
      v8f acc0 = {0.f, 0.f, 0.f, 0.f, 0.f, 0.f, 0.f, 0.f};
      v8f acc1 = {0.f, 0.f, 0.f, 0.f, 0.f, 0.f, 0.f, 0.f};
#pragma unroll
      for (int kk = 0; kk < 16; ++kk) {
        acc0 = __builtin_amdgcn_wmma_f32_16x16x4_f32(
            false, a[kk], false, b0[kk], (short)0, acc0, false, false);
        acc1 = __builtin_amdgcn_wmma_f32_16x16x4_f32(
            false, a[kk], false, b1[kk], (short)0, acc1, false, false);
      }

      const int   code0 = codeBase + (t + 0) * 16 + r;
      const int   code1 = codeBase + (t + 1) * 16 + r;
      const float cc0 = cAll[code0];
      const float cc1 = cAll[code1];
#pragma unroll
      for (int v = 0; v < 8; ++v) {
        float v0 = acc0[v] - cc0;   // <l,e> - 0.5||e||^2; argmax == argmin dist
        if (v0 > bestVal[v]) { bestVal[v] = v0; bestIdx[v] = code0; }
        float v1 = acc1[v] - cc1;   // checked second: earlier code wins ties
        if (v1 > bestVal[v]) { bestVal[v] = v1; bestIdx[v] = code1; }
      }
    }

    wait_async_all();                // my async writes to the other buffer done
    __syncthreads();                 // everyone done reading cur / staging next
  }

  // ---- cross-lane argmax within each 16-lane half (row m = v + 8p) ----
#pragma unroll
  for (int v = 0; v < 8; ++v) {
    float bv = bestVal[v]; int bi = bestIdx[v];
#pragma unroll
    for (int off = 1; off < 16; off <<= 1) {
      float ov = __shfl_xor(bv, off, 32);
      int   oi = __shfl_xor(bi, off, 32);
      if (ov > bv || (ov == bv && oi < bi)) { bv = ov; bi = oi; }
    }
    if (r == 0) idxS[wave * ROWS_PER_WAVE + (v + 8 * p)] = bi;
  }
  __syncthreads();

  // ---- gather winning codebook rows, write quantized (== straight-through), SSE ----
  float sse = 0.f;
#pragma unroll
  for (int m = 0; m < ROWS_PER_WAVE; ++m) {
    const int idx = idxS[wave * ROWS_PER_WAVE + m];
    const int row = rowBase + m;
    v2f q = *(const v2f*)(embedding + idx * DIM + 2 * lane);
    v2f l = *(const v2f*)(latents  + row * DIM + 2 * lane);
    float dx = q.x - l.x, dy = q.y - l.y;
    sse += dx * dx + dy * dy;
    *(v2f*)(out + row * DIM + 2 * lane) = q;
  }
#pragma unroll
  for (int off = 1; off < 32; off <<= 1)
    sse += __shfl_xor(sse, off, 32);
  if (lane == 0) sseS[wave] = sse;
  __syncthreads();
  if (tid == 0) {
    float s = 0.f;
#pragma unroll
    for (int w = 0; w < WAVES; ++w) s += sseS[w];
    blockSse[blockIdx.x] = s;        // every block writes its slot: no zero-init needed
  }
}

__global__ void vq_loss_kernel(const float* __restrict__ blockSse,
                               float* __restrict__ out, int nBlocks)
{
  if (threadIdx.x == 0 && blockIdx.x == 0) {
    float s = 0.f;
    for (int i = 0; i < nBlocks; ++i) s += blockSse[i];   // fixed order: deterministic
    // commitment*beta + embedding == (1+beta) * mean((q-l)^2)
    out[N_TOK * DIM] = s * (1.0f + BETA) / (float)(N_TOK * DIM);
  }
}

extern "C" void kernel_launch(void* const* d_in, const int* in_sizes, int n_in,
                              void* d_out, int out_size, void* d_ws, size_t ws_size,
                              hipStream_t stream) {
  const float* latents   = (const float*)d_in[0];   // [65536, 64] f32
  const float* embedding = (const float*)d_in[1];   // [1024, 64] f32
  float* out      = (float*)d_out;                  // [65536*64] quantized + [1] loss
  float* blockSse = (float*)d_ws;                   // 512 floats of partial SSE

  const int nBlocks = N_TOK / ROWS_PER_BLOCK;       // 512
  vq_argmin_kernel<<<nBlocks, 256, 0, stream>>>(latents, embedding, out, blockSse);
  vq_loss_kernel<<<1, 32, 0, stream>>>(blockSse, out, nBlocks);
}